// Ohem_15934328668325
// MI455X (gfx1250) — compile-verified
//
#include <hip/hip_runtime.h>

// ---------------- problem constants (from reference) ----------------
constexpr int      CLS    = 19;
constexpr unsigned NN     = 8;
constexpr unsigned HH     = 512;
constexpr unsigned WW     = 1024;
constexpr unsigned HW     = HH * WW;        // 524288 = 2^19
constexpr unsigned P      = NN * HW;        // 4194304
constexpr unsigned IGNORE = 255u;
constexpr unsigned MINKEPT= 100000u;
constexpr unsigned INF_BITS = 0x7F800000u;

// ---------------- workspace layout ----------------
struct Ctrl {
  unsigned counts[CLS];
  unsigned num_valid;
  float    weight[CLS];
  unsigned prefix;
  unsigned krem;
  float    threshold;
  unsigned do_ohem;
};
constexpr size_t OFF_H2048 = 1024;                    // 2048 * 4 = 8 KB
constexpr size_t OFF_H1024 = OFF_H2048 + 8192;        // 1024 * 4 = 4 KB
constexpr size_t OFF_PART  = 16384;                   // 1024 blocks * 2 doubles = 16 KB
constexpr size_t OFF_BITS  = 65536;                   // P * 4 = 16 MB
constexpr size_t OFF_LOGP  = OFF_BITS + (size_t)P * 4;
constexpr int    RED_BLOCKS = 1024;

// ---------------- CDNA5 async global->LDS path (guarded) ----------------
#if defined(__has_builtin)
#if __has_builtin(__builtin_amdgcn_global_load_async_to_lds_b128) && \
    __has_builtin(__builtin_amdgcn_s_wait_asynccnt)
#define USE_ASYNC_LDS 1
#endif
#endif

typedef int   v4i __attribute__((ext_vector_type(4)));
typedef float v4f __attribute__((ext_vector_type(4)));
typedef __attribute__((address_space(1))) v4i* gptr_v4i;
typedef __attribute__((address_space(3))) v4i* lptr_v4i;

__device__ __forceinline__ void async_copy16(const void* g, void* l) {
#if defined(USE_ASYNC_LDS)
  __builtin_amdgcn_global_load_async_to_lds_b128((gptr_v4i)(g), (lptr_v4i)(l), 0, 0);
#else
  (void)g; (void)l;
#endif
}
#if defined(USE_ASYNC_LDS)
#define ASYNC_WAIT(n) __builtin_amdgcn_s_wait_asynccnt(n)
#else
#define ASYNC_WAIT(n)
#endif

// ---------------- K0: zero control + histograms + partials ----------------
__global__ void __launch_bounds__(256) zero_kernel(unsigned* w) {
  w[blockIdx.x * 256u + threadIdx.x] = 0u;   // 64 blocks -> 64 KB
}

// ---------------- K1: class counts + num_valid ----------------
__global__ void __launch_bounds__(256) count_kernel(const int* __restrict__ target,
                                                    Ctrl* __restrict__ ctrl) {
  __shared__ unsigned lc[CLS + 1];
  const int tid = threadIdx.x;
  if (tid < CLS + 1) lc[tid] = 0u;
  __syncthreads();
  for (unsigned i = blockIdx.x * 256u + tid; i < P; i += gridDim.x * 256u) {
    const int lab = target[i];
    if ((unsigned)lab < (unsigned)CLS) atomicAdd(&lc[lab], 1u);
    const unsigned long long b = __ballot(lab != (int)IGNORE);
    if ((tid & (warpSize - 1)) == 0) atomicAdd(&lc[CLS], (unsigned)__popcll(b));
  }
  __syncthreads();
  if (tid < CLS)       { if (lc[tid]) atomicAdd(&ctrl->counts[tid], lc[tid]); }
  else if (tid == CLS) { if (lc[CLS]) atomicAdd(&ctrl->num_valid, lc[CLS]); }
}

// ---------------- K2: median-based class weights ----------------
__global__ void weights_kernel(Ctrl* __restrict__ ctrl) {
  if (threadIdx.x != 0 || blockIdx.x != 0) return;
  float cnt[CLS], vals[CLS];
  int m = 0;
  for (int c = 0; c < CLS; ++c) {
    cnt[c] = (float)ctrl->counts[c];
    if (cnt[c] > 0.f) vals[m++] = cnt[c];
  }
  float med = 1.0f;
  if (m > 0) {
    for (int i = 1; i < m; ++i) {           // insertion sort (<=19 elems)
      float v = vals[i]; int j = i - 1;
      while (j >= 0 && vals[j] > v) { vals[j + 1] = vals[j]; --j; }
      vals[j + 1] = v;
    }
    med = (m & 1) ? vals[m / 2] : 0.5f * (vals[m / 2 - 1] + vals[m / 2]);
  }
  for (int c = 0; c < CLS; ++c)
    ctrl->weight[c] = (cnt[c] > 0.f) ? med / fmaxf(cnt[c], 1.0f) : 1.0f;
}

// ---------------- K3: streaming log-softmax + fused level-0 histogram ----------
// 256 threads * 4 pixels = 1024-pixel tile (divides HW, never crosses an image).
// predict is read exactly once -> non-temporal loads keep the 192 MB L2 free for
// the bits/logp/target working set (~50 MB) re-read by the later passes.
__global__ void __launch_bounds__(256) main_pass_kernel(
    const float* __restrict__ predict, const int* __restrict__ target,
    unsigned* __restrict__ outbits, float* __restrict__ outlogp,
    unsigned* __restrict__ hist2048) {
  __shared__ unsigned lh[2048];
  const int tid = threadIdx.x;
  for (int j = tid; j < 2048; j += 256) lh[j] = 0u;
  __syncthreads();

  const unsigned p0  = blockIdx.x * 1024u + (unsigned)tid * 4u;
  const unsigned n   = p0 >> 19;            // / HW
  const unsigned off = p0 & (HW - 1u);
  const float* img = predict + (size_t)n * CLS * HW + off;

  v4f x[CLS];
#pragma unroll
  for (int c = 0; c < CLS; ++c)
    x[c] = __builtin_nontemporal_load(
        reinterpret_cast<const v4f*>(img + (size_t)c * HW));
  const int4 lab = *reinterpret_cast<const int4*>(target + p0);

  v4f mx = x[0];
#pragma unroll
  for (int c = 1; c < CLS; ++c) {
    mx.x = fmaxf(mx.x, x[c].x); mx.y = fmaxf(mx.y, x[c].y);
    mx.z = fmaxf(mx.z, x[c].z); mx.w = fmaxf(mx.w, x[c].w);
  }
  v4f xt = x[0];                             // safe_lab default = class 0
  float sx = 0.f, sy = 0.f, sz = 0.f, sw = 0.f;
#pragma unroll
  for (int c = 0; c < CLS; ++c) {
    const v4f v = x[c];
    sx += expf(v.x - mx.x); sy += expf(v.y - mx.y);
    sz += expf(v.z - mx.z); sw += expf(v.w - mx.w);
    if (lab.x == c) xt.x = v.x;
    if (lab.y == c) xt.y = v.y;
    if (lab.z == c) xt.z = v.z;
    if (lab.w == c) xt.w = v.w;
  }
  float4 lp;
  lp.x = xt.x - mx.x - logf(sx); lp.y = xt.y - mx.y - logf(sy);
  lp.z = xt.z - mx.z - logf(sz); lp.w = xt.w - mx.w - logf(sw);
  *reinterpret_cast<float4*>(outlogp + p0) = lp;

  uint4 ub;
  ub.x = (lab.x != (int)IGNORE) ? __float_as_uint(expf(lp.x)) : INF_BITS;
  ub.y = (lab.y != (int)IGNORE) ? __float_as_uint(expf(lp.y)) : INF_BITS;
  ub.z = (lab.z != (int)IGNORE) ? __float_as_uint(expf(lp.z)) : INF_BITS;
  ub.w = (lab.w != (int)IGNORE) ? __float_as_uint(expf(lp.w)) : INF_BITS;
  *reinterpret_cast<uint4*>(outbits + p0) = ub;

  atomicAdd(&lh[ub.x >> 20], 1u); atomicAdd(&lh[ub.y >> 20], 1u);
  atomicAdd(&lh[ub.z >> 20], 1u); atomicAdd(&lh[ub.w >> 20], 1u);
  __syncthreads();
  for (int j = tid; j < 2048; j += 256) {
    const unsigned v = lh[j];
    if (v) atomicAdd(&hist2048[j], v);
  }
}

// ---------------- K5/K7: 1024-bin radix histogram pass (async double-buffered) --
__global__ void __launch_bounds__(256) radix_hist_kernel(
    const unsigned* __restrict__ bits, const Ctrl* __restrict__ ctrl,
    unsigned* __restrict__ hist, unsigned pmask, int shift) {
  __shared__ unsigned lh[1024];
  const int tid = threadIdx.x;
  for (int j = tid; j < 1024; j += 256) lh[j] = 0u;
  __syncthreads();
  const unsigned pfx = ctrl->prefix & pmask;
  const int nchunks = (int)(P / 1024u);     // 4096 chunks of 1024 u32 (4 KB)
  const int stride  = gridDim.x;

#if defined(USE_ASYNC_LDS)
  __shared__ unsigned stage[2][1024];       // 2 x 4 KB, each lane owns 16 B
  const int c0 = blockIdx.x;
  async_copy16(bits + (size_t)c0 * 1024u + tid * 4, &stage[0][tid * 4]);
  int i = 0;
  for (int c = c0; c < nchunks; c += stride, ++i) {
    const int cn = c + stride;
    if (cn < nchunks) {
      async_copy16(bits + (size_t)cn * 1024u + tid * 4, &stage[(i + 1) & 1][tid * 4]);
      ASYNC_WAIT(1);                        // current buffer complete (in-order)
    } else {
      ASYNC_WAIT(0);
    }
    const unsigned* s = &stage[i & 1][tid * 4];  // each thread reads only its own 16 B
#pragma unroll
    for (int j = 0; j < 4; ++j) {
      const unsigned u = s[j];
      if ((u & pmask) == pfx) atomicAdd(&lh[(u >> shift) & 1023u], 1u);
    }
  }
#else
  for (int c = blockIdx.x; c < nchunks; c += stride) {
    const uint4 v = *reinterpret_cast<const uint4*>(bits + (size_t)c * 1024u + tid * 4);
    if ((v.x & pmask) == pfx) atomicAdd(&lh[(v.x >> shift) & 1023u], 1u);
    if ((v.y & pmask) == pfx) atomicAdd(&lh[(v.y >> shift) & 1023u], 1u);
    if ((v.z & pmask) == pfx) atomicAdd(&lh[(v.z >> shift) & 1023u], 1u);
    if ((v.w & pmask) == pfx) atomicAdd(&lh[(v.w >> shift) & 1023u], 1u);
  }
#endif
  __syncthreads();
  for (int j = tid; j < 1024; j += 256) {
    const unsigned v = lh[j];
    if (v) atomicAdd(&hist[j], v);
  }
}

// ---------------- K4/K6/K8: radix-select scan ----------------
// mode 0: init (compute k from num_valid), scan hist2048 (shift 20)
// mode 1: intermediate, scan hist1024 (shift 10) and re-zero it for the next pass
// mode 2: final, scan hist1024 (shift 0) -> exact kth value -> threshold
__global__ void __launch_bounds__(256) scan_kernel(Ctrl* __restrict__ ctrl,
                                                   unsigned* __restrict__ hist,
                                                   int nbins, int shift, int mode) {
  const int tid = threadIdx.x;
  if (tid == 0) {
    long long k;
    if (mode == 0) {
      const unsigned nv = ctrl->num_valid;
      const unsigned kq = nv < MINKEPT ? nv : MINKEPT;
      k = (long long)kq - 1; if (k < 0) k = 0;             // max(k,0)
      ctrl->do_ohem = (nv > MINKEPT) ? 1u : 0u;
    } else {
      k = (long long)ctrl->krem;
    }
    unsigned long long cum = 0;
    int b = 0;
    for (; b < nbins; ++b) {
      const unsigned c = hist[b];
      if (cum + c > (unsigned long long)k) break;
      cum += c;
    }
    if (b >= nbins) b = nbins - 1;                         // safety (never hit)
    const unsigned newpfx = ((mode == 0) ? 0u : ctrl->prefix) | ((unsigned)b << shift);
    ctrl->prefix = newpfx;
    ctrl->krem   = (unsigned)((unsigned long long)k - cum);
    if (mode == 2) {
      const float kth = __uint_as_float(newpfx);           // exact kth of our bits
      ctrl->threshold = (kth > 0.7f) ? kth : 0.7f;
    }
  }
  __syncthreads();
  if (mode == 1)
    for (int j = tid; j < nbins; j += 256) hist[j] = 0u;   // reuse buffer in pass C
}

// ---------------- K9: deterministic weighted reduction ----------------
__global__ void __launch_bounds__(256) reduce_kernel(
    const int* __restrict__ target, const unsigned* __restrict__ bits,
    const float* __restrict__ logp, const Ctrl* __restrict__ ctrl,
    double* __restrict__ partials) {
  __shared__ float wgt[CLS];
  const int tid = threadIdx.x;
  if (tid < CLS) wgt[tid] = ctrl->weight[tid];
  __syncthreads();
  const float    th  = ctrl->threshold;
  const unsigned doh = ctrl->do_ohem;
  float lsum = 0.f, wsum = 0.f;
  for (unsigned i = blockIdx.x * 256u + tid; i < P; i += (unsigned)RED_BLOCKS * 256u) {
    const int lab = target[i];
    const bool valid = (lab != (int)IGNORE);
    const unsigned u = bits[i];
    const bool kept = doh ? (valid && (__uint_as_float(u) <= th)) : valid;
    if (kept) {
      unsigned sl = (unsigned)lab; if (sl >= (unsigned)CLS) sl = 0u;
      const float w = wgt[sl];
      wsum += w;
      lsum += w * (-logp[i]);
    }
  }
  __shared__ float r0[256], r1[256];
  r0[tid] = lsum; r1[tid] = wsum;
  __syncthreads();
  for (int s = 128; s > 0; s >>= 1) {
    if (tid < s) { r0[tid] += r0[tid + s]; r1[tid] += r1[tid + s]; }
    __syncthreads();
  }
  if (tid == 0) {
    partials[2 * blockIdx.x]     = (double)r0[0];
    partials[2 * blockIdx.x + 1] = (double)r1[0];
  }
}

// ---------------- K10: final (fixed-order, deterministic) ----------------
__global__ void finalize_kernel(const double* __restrict__ partials, float* __restrict__ out) {
  if (threadIdx.x == 0 && blockIdx.x == 0) {
    double ls = 0.0, ws = 0.0;
    for (int i = 0; i < RED_BLOCKS; ++i) { ls += partials[2 * i]; ws += partials[2 * i + 1]; }
    const double denom = (ws > 1e-12) ? ws : 1e-12;
    out[0] = (float)(ls / denom);
  }
}

// ---------------- host launch ----------------
extern "C" void kernel_launch(void* const* d_in, const int* in_sizes, int n_in,
                              void* d_out, int out_size, void* d_ws, size_t ws_size,
                              hipStream_t stream) {
  (void)in_sizes; (void)n_in; (void)out_size; (void)ws_size;
  const float* predict = (const float*)d_in[0];
  const int*   target  = (const int*)d_in[1];
  float*       out     = (float*)d_out;
  char*        ws      = (char*)d_ws;

  Ctrl*     ctrl     = (Ctrl*)ws;
  unsigned* hist2048 = (unsigned*)(ws + OFF_H2048);
  unsigned* hist1024 = (unsigned*)(ws + OFF_H1024);
  double*   partials = (double*)(ws + OFF_PART);
  unsigned* bits     = (unsigned*)(ws + OFF_BITS);
  float*    logp     = (float*)(ws + OFF_LOGP);

  zero_kernel<<<64, 256, 0, stream>>>((unsigned*)ws);                       // 64 KB
  count_kernel<<<1024, 256, 0, stream>>>(target, ctrl);
  weights_kernel<<<1, 32, 0, stream>>>(ctrl);
  main_pass_kernel<<<P / 1024, 256, 0, stream>>>(predict, target, bits, logp, hist2048);
  scan_kernel<<<1, 256, 0, stream>>>(ctrl, hist2048, 2048, 20, 0);
  radix_hist_kernel<<<256, 256, 0, stream>>>(bits, ctrl, hist1024, 0xFFF00000u, 10);
  scan_kernel<<<1, 256, 0, stream>>>(ctrl, hist1024, 1024, 10, 1);
  radix_hist_kernel<<<256, 256, 0, stream>>>(bits, ctrl, hist1024, 0xFFFFFC00u, 0);
  scan_kernel<<<1, 256, 0, stream>>>(ctrl, hist1024, 1024, 0, 2);
  reduce_kernel<<<RED_BLOCKS, 256, 0, stream>>>(target, bits, logp, ctrl, partials);
  finalize_kernel<<<1, 32, 0, stream>>>(partials, out);
}